// SequentialSystem_3092376453578
// MI455X (gfx1250) — compile-verified
//
#include <hip/hip_runtime.h>
#include <cstdint>

#define TPB 256
#define RT_EPS 1e-6f
#define TRANSMIT 0.99f

// ---- CDNA5 async global<->LDS movers (gfx1250, ASYNCcnt-tracked) ----------

__device__ __forceinline__ unsigned lds_off(const void* p) {
    // addrspace(3) offsets live in the low 32 bits of the generic pointer
    return (unsigned)(uintptr_t)p;
}

__device__ __forceinline__ void async_g2lds_b32(const void* gbase, unsigned gByteOff,
                                                unsigned ldsByteOff) {
    asm volatile("global_load_async_to_lds_b32 %0, %1, %2 th:TH_LOAD_NT"
                 :
                 : "v"(ldsByteOff), "v"(gByteOff), "s"(gbase)
                 : "memory");
}

__device__ __forceinline__ void async_g2lds_b128(const void* gbase, unsigned gByteOff,
                                                 unsigned ldsByteOff) {
    asm volatile("global_load_async_to_lds_b128 %0, %1, %2 th:TH_LOAD_NT"
                 :
                 : "v"(ldsByteOff), "v"(gByteOff), "s"(gbase)
                 : "memory");
}

__device__ __forceinline__ void async_lds2g_b32(void* gbase, unsigned gByteOff,
                                                unsigned ldsByteOff) {
    asm volatile("global_store_async_from_lds_b32 %0, %1, %2 th:TH_STORE_NT"
                 :
                 : "v"(gByteOff), "v"(ldsByteOff), "s"(gbase)
                 : "memory");
}

__device__ __forceinline__ void async_lds2g_b128(void* gbase, unsigned gByteOff,
                                                 unsigned ldsByteOff) {
    asm volatile("global_store_async_from_lds_b128 %0, %1, %2 th:TH_STORE_NT"
                 :
                 : "v"(gByteOff), "v"(ldsByteOff), "s"(gbase)
                 : "memory");
}

__device__ __forceinline__ void wait_asynccnt0() {
    asm volatile("s_wait_asynccnt 0" ::: "memory");
}

// ---- kernel ---------------------------------------------------------------

__global__ __launch_bounds__(TPB) void raytrace8_kernel(
    const float* __restrict__ pos,   // [N,3]
    const float* __restrict__ dirn,  // [N,3]
    const float* __restrict__ inten, // [N]
    const float* __restrict__ sz,    // [8]
    const float* __restrict__ sc,    // [8]
    const float* __restrict__ sap,   // [8]
    const float* __restrict__ seta,  // [8]
    float* __restrict__ out,         // [N,7]
    int n)
{
    // shIn layout (dwords): [0,768) pos block | [768,1536) dir block | [1536,1792) intensity
    __shared__ __align__(16) float shIn[TPB * 7];
    __shared__ __align__(16) float shOut[TPB * 7];

    const int tid = threadIdx.x;
    const unsigned r0 = (unsigned)blockIdx.x * TPB;
    const unsigned un = (unsigned)n;
    const bool fullBlock = (r0 + TPB) <= un;

    const unsigned ldsInBase = lds_off(&shIn[0]);

    // ---- stage inputs into LDS with async copies ----
    if (fullBlock) {
        // contiguous 16B-aligned regions: use b128 movers (wave-uniform guards:
        // 192 and 64 are multiples of 32)
        const unsigned pdByte = r0 * 12u;             // pos/dir block byte offset
        const unsigned t16 = (unsigned)tid * 16u;
        if (tid < 192) {                              // 192 * 16B = 3072B each
            async_g2lds_b128(pos,  pdByte + t16, ldsInBase + t16);
            async_g2lds_b128(dirn, pdByte + t16, ldsInBase + 3072u + t16);
        }
        if (tid < 64) {                               // 64 * 16B = 1024B
            async_g2lds_b128(inten, r0 * 4u + t16, ldsInBase + 6144u + t16);
        }
    } else {
        // tail block: guarded b32 path
        const unsigned nPosF = 3u * un;
#pragma unroll
        for (int k = 0; k < 3; ++k) {
            unsigned gi = r0 * 3u + (unsigned)tid + (unsigned)(k * TPB);
            if (gi < nPosF) {
                async_g2lds_b32(pos,  gi * 4u, ldsInBase + (tid + k * TPB) * 4u);
                async_g2lds_b32(dirn, gi * 4u, ldsInBase + 3072u + (tid + k * TPB) * 4u);
            }
        }
        unsigned gi = r0 + (unsigned)tid;
        if (gi < un)
            async_g2lds_b32(inten, gi * 4u, ldsInBase + 6144u + (unsigned)tid * 4u);
    }
    wait_asynccnt0();
    __syncthreads();

    // ---- per-ray state (stride 3/7 dword LDS reads: conflict-free) ----
    float px = shIn[tid * 3 + 0];
    float py = shIn[tid * 3 + 1];
    float pz = shIn[tid * 3 + 2];
    float dx = shIn[768 + tid * 3 + 0];
    float dy = shIn[768 + tid * 3 + 1];
    float dz = shIn[768 + tid * 3 + 2];
    float I  = shIn[1536 + tid];

    // ---- 8 surfaces, fully unrolled; params are wave-uniform (s_load) ----
#pragma unroll
    for (int i = 0; i < 8; ++i) {
        const float zv  = sz[i];
        const float c   = sc[i];
        const float ap  = sap[i];
        const float eta = seta[i];

        const bool  is_plane = fabsf(c) < 1e-12f;
        const float c_safe   = is_plane ? 1.0f : c;
        const float Rs       = 1.0f / c_safe;
        const float cz       = zv + Rs;               // sphere center z

        // sphere intersection (|d| = 1)
        const float ocx = px, ocy = py, ocz = pz - cz;
        const float b   = dx * ocx + dy * ocy + dz * ocz;
        const float cq  = ocx * ocx + ocy * ocy + ocz * ocz - Rs * Rs;
        const float disc = b * b - cq;
        const bool  valid_disc = disc >= 0.0f;
        const float sq = __builtin_sqrtf(valid_disc ? disc : 0.0f);
        const float t0 = -b - sq;
        const float t1 = -b + sq;
        const float t_sph = (t0 > RT_EPS) ? t0 : t1;

        // plane fallback
        const float dz_safe = (fabsf(dz) > RT_EPS) ? dz : 1.0f;
        const float t_pl    = (zv - pz) / dz_safe;
        const float t       = is_plane ? t_pl : t_sph;

        const float hx = px + t * dx;
        const float hy = py + t * dy;
        const float hz = pz + t * dz;
        const float r2 = hx * hx + hy * hy;

        bool mask = (t > RT_EPS) && (r2 <= ap * ap) && (is_plane || valid_disc);

        // oriented surface normal
        float nx = hx * c_safe;
        float ny = hy * c_safe;
        float nz = (hz - cz) * c_safe;
        if (is_plane) { nx = 0.0f; ny = 0.0f; nz = 1.0f; }
        const float cosdn = dx * nx + dy * ny + dz * nz;
        if (cosdn > 0.0f) { nx = -nx; ny = -ny; nz = -nz; }

        // Snell refraction (vector form)
        const float cos_i  = -(dx * nx + dy * ny + dz * nz);
        const float sin2_t = eta * eta * (1.0f - cos_i * cos_i);
        const bool  tir    = sin2_t > 1.0f;
        mask = mask && (!tir);
        const float cos_t = __builtin_sqrtf(tir ? 0.0f : (1.0f - sin2_t));
        const float kk    = eta * cos_i - cos_t;
        float ndx = eta * dx + kk * nx;
        float ndy = eta * dy + kk * ny;
        float ndz = eta * dz + kk * nz;
        const float nn  = fmaxf(ndx * ndx + ndy * ndy + ndz * ndz, RT_EPS);
        const float inv = 1.0f / __builtin_sqrtf(nn);
        ndx *= inv; ndy *= inv; ndz *= inv;

        if (mask) {
            px = hx; py = hy; pz = hz;
            dx = ndx; dy = ndy; dz = ndz;
            I *= TRANSMIT;
        }
    }

    // ---- stage outputs (stride-7 LDS writes: conflict-free) ----
    shOut[tid * 7 + 0] = px;
    shOut[tid * 7 + 1] = py;
    shOut[tid * 7 + 2] = pz;
    shOut[tid * 7 + 3] = dx;
    shOut[tid * 7 + 4] = dy;
    shOut[tid * 7 + 5] = dz;
    shOut[tid * 7 + 6] = I;
    __syncthreads();

    // ---- async coalesced store LDS -> [N,7] global ----
    const unsigned ldsOutBase = lds_off(&shOut[0]);
    if (fullBlock) {
        // 448 x b128 per block; second pass guard (192) is wave-aligned
        const unsigned outByte = r0 * 28u;
        const unsigned t16 = (unsigned)tid * 16u;
        async_lds2g_b128(out, outByte + t16, ldsOutBase + t16);
        if (tid < 192) {
            const unsigned o2 = t16 + 4096u;          // (tid+256)*16
            async_lds2g_b128(out, outByte + o2, ldsOutBase + o2);
        }
    } else {
        const unsigned nOutF = 7u * un;
#pragma unroll
        for (int k = 0; k < 7; ++k) {
            unsigned gi = r0 * 7u + (unsigned)tid + (unsigned)(k * TPB);
            if (gi < nOutF)
                async_lds2g_b32(out, gi * 4u, ldsOutBase + (tid + k * TPB) * 4u);
        }
    }
    wait_asynccnt0();  // S_ENDPGM also waits-idle; belt & braces
}

// ---- launcher -------------------------------------------------------------

extern "C" void kernel_launch(void* const* d_in, const int* in_sizes, int n_in,
                              void* d_out, int out_size, void* d_ws, size_t ws_size,
                              hipStream_t stream) {
    const float* pos   = (const float*)d_in[0];
    const float* dirn  = (const float*)d_in[1];
    const float* inten = (const float*)d_in[2];
    const float* sz    = (const float*)d_in[3];
    const float* sc    = (const float*)d_in[4];
    const float* sap   = (const float*)d_in[5];
    const float* seta  = (const float*)d_in[6];
    float* out = (float*)d_out;

    const int n = in_sizes[2];                 // intensity count == N_RAYS
    const int blocks = (n + TPB - 1) / TPB;

    raytrace8_kernel<<<dim3(blocks), dim3(TPB), 0, stream>>>(
        pos, dirn, inten, sz, sc, sap, seta, out, n);
}